// PolNetLSTM_24421184045755
// MI455X (gfx1250) — compile-verified
//
#include <hip/hip_runtime.h>
#include <math.h>

// ---------------------------------------------------------------------------
// PolNetLSTM forward on gfx1250 (MI455X), fp32 via V_WMMA_F32_16X16X4_F32,
// with async global->LDS weight staging (GLOBAL_LOAD_ASYNC_TO_LDS_B128,
// ASYNCcnt double-buffering).
//   T=512, B=256, OBS=64, ACT=8, H=1024, C=512, 4C=2048
//
// Structure (chunked over time, chunk size Tc chosen from ws_size):
//   init     : h0/c0 -> ws, broadcast log_std into d_out
//   per chunk:
//     embed  : x_emb = relu(xs @ W_in^T + b_in)            [Tc*B, H]    (WMMA)
//     gx     : gx = x_emb @ W_ih^T + b_ih                  [Tc*B, 4C]   (WMMA,
//              async-staged B tile in LDS, double buffered)
//     Tc x step: gates = gx[t] + (h*keep) @ W_hh^T + b_hh; LSTM update  (WMMA,
//              4 gate weight tiles async-staged in LDS, double buffered)
//     mean   : means = tanh(hiddens @ W_mean^T + b_mean)   (VALU, N=8)
//   final    : h_T, c_T -> d_out
// ---------------------------------------------------------------------------

#define T_STEPS 512
#define BATCH   256
#define OBSD    64
#define ACTD    8
#define HDIM    1024
#define CDIM    512
#define GDIM    (4 * CDIM)
#define KC      64

typedef __attribute__((ext_vector_type(2))) float v2f;
typedef __attribute__((ext_vector_type(8))) float v8f;

// D = A(16x4,f32) * B(4x16,f32) + C(16x16,f32)   -- wave32 matrix op
__device__ __forceinline__ v8f wmma_f32(v2f a, v2f b, v8f c) {
    return __builtin_amdgcn_wmma_f32_16x16x4_f32(
        false, a, false, b, (short)0, c, false, false);
}

// A fragment: A row-major [M x K]. ISA 16x4 f32 A layout:
//   lanes 0-15: row m0+lane, cols {k0,k0+1}; lanes 16-31: row m0+lane-16, {k0+2,k0+3}
__device__ __forceinline__ v2f load_a(const float* __restrict__ A, int lda,
                                      int m0, int k0, int lane) {
    int row = m0 + (lane & 15);
    int col = k0 + ((lane >> 4) << 1);
    return *reinterpret_cast<const v2f*>(A + (size_t)row * lda + col);
}

// B fragment straight from W[N][K] row-major (B = W^T): contiguous float2.
__device__ __forceinline__ v2f load_bT(const float* __restrict__ W, int ldw,
                                       int n0, int k0, int lane) {
    int n = n0 + (lane & 15);
    int k = k0 + ((lane >> 4) << 1);
    return *reinterpret_cast<const v2f*>(W + (size_t)n * ldw + k);
}

// Async 16-byte global -> LDS copy (per lane), tracked by ASYNCcnt.
// Low 32 bits of a generic pointer to __shared__ == wave-relative LDS offset.
__device__ __forceinline__ void async_b128(const float* gsrc, float* ldst) {
    asm volatile("global_load_async_to_lds_b128 %0, %1, off"
                 :: "v"((unsigned int)(size_t)ldst), "v"(gsrc)
                 : "memory");
}
// Wait until at most N async ops remain in flight (in-order completion).
#define ASYNC_WAIT(N) asm volatile("s_wait_asynccnt " #N ::: "memory")

__device__ __forceinline__ float sigf(float x) { return 1.0f / (1.0f + __expf(-x)); }

// ---------------------------------------------------------------------------
__global__ void init_kernel(const float* __restrict__ h0, const float* __restrict__ c0,
                            const float* __restrict__ log_std_param,
                            float* __restrict__ hcarry, float* __restrict__ cbuf,
                            float* __restrict__ out_logstd) {
    int idx = blockIdx.x * blockDim.x + threadIdx.x;
    if (idx < BATCH * CDIM) { hcarry[idx] = h0[idx]; cbuf[idx] = c0[idx]; }
    if (idx < T_STEPS * BATCH * ACTD) out_logstd[idx] = log_std_param[idx & (ACTD - 1)];
}

// ---------------------------------------------------------------------------
// embed: x_emb = relu(xs @ W_in^T + b_in), M rows (M = Tc*B), K=64, N=1024.
// 4 waves/block, one 16x16 tile each; grid (H/64, M/16). K is tiny: direct loads.
// ---------------------------------------------------------------------------
__global__ void __launch_bounds__(128)
embed_kernel(const float* __restrict__ xs_c,   // [M, OBS]
             const float* __restrict__ W_in,   // [H, OBS]
             const float* __restrict__ b_in,   // [H]
             float* __restrict__ x_emb) {      // [M, H]
    const int wave = threadIdx.x >> 5;
    const int lane = threadIdx.x & 31;
    const int m0 = blockIdx.y * 16;
    const int n0 = blockIdx.x * 64 + wave * 16;

    v8f acc = (v8f){0.f,0.f,0.f,0.f,0.f,0.f,0.f,0.f};
#pragma unroll
    for (int k0 = 0; k0 < OBSD; k0 += 4) {
        v2f a = load_a(xs_c, OBSD, m0, k0, lane);
        v2f b = load_bT(W_in, OBSD, n0, k0, lane);
        acc = wmma_f32(a, b, acc);
    }
    const int n = n0 + (lane & 15);
    const int mbase = m0 + 8 * (lane >> 4);
    const float bias = b_in[n];
#pragma unroll
    for (int r = 0; r < 8; ++r) {
        float v = acc[r] + bias;
        x_emb[(size_t)(mbase + r) * HDIM + n] = v > 0.f ? v : 0.f;
    }
}

// ---------------------------------------------------------------------------
// gx: gx = x_emb @ W_ih^T + b_ih.  M rows, K=1024, N=2048.
// Block = 4 waves sharing one n-tile (16 cols); wave tile = 64 rows x 16 cols
// (4 accumulators). W_ih tile [16 x 64k] async-staged into LDS, double
// buffered (stride 68 -> banks 4n+k, conflict-free ds_load_b64 reads).
// ---------------------------------------------------------------------------
__global__ void __launch_bounds__(128)
gx_kernel(const float* __restrict__ x_emb,  // [M, H]
          const float* __restrict__ W_ih,   // [4C, H]
          const float* __restrict__ b_ih,   // [4C]
          float* __restrict__ gx) {         // [M, 4C]
    __shared__ float lb[2][16][KC + 4];
    const int wave = threadIdx.x >> 5;
    const int lane = threadIdx.x & 31;
    const int n0 = blockIdx.x * 16;
    const int m0 = blockIdx.y * 256 + wave * 64;

    // staging map: 16x64 tile = 1024 floats / 128 thr = 8 floats (2 x b128) each
    const int nrow = threadIdx.x >> 3;
    const int kcol = (threadIdx.x & 7) * 8;
    const float* wsrc = W_ih + (size_t)(n0 + nrow) * HDIM + kcol;
    float* ld[2] = { &lb[0][nrow][kcol], &lb[1][nrow][kcol] };

    v8f acc[4];
#pragma unroll
    for (int s = 0; s < 4; ++s) acc[s] = (v8f){0.f,0.f,0.f,0.f,0.f,0.f,0.f,0.f};

    // prologue: stage chunk 0 into buffer 0
    async_b128(wsrc + 0, ld[0]);
    async_b128(wsrc + 4, ld[0] + 4);

    const int NCH = HDIM / KC;   // 16
    for (int ci = 0; ci < NCH; ++ci) {
        const int k0 = ci * KC;
        __syncthreads();                       // prev buffer free for refill
        if (ci + 1 < NCH) {
            const float* s = wsrc + (size_t)(ci + 1) * KC;
            float* d = ld[(ci + 1) & 1];
            async_b128(s + 0, d);
            async_b128(s + 4, d + 4);
            ASYNC_WAIT(0x2);                   // current buffer complete; next in flight
        } else {
            ASYNC_WAIT(0x0);
        }
        __syncthreads();                       // all waves see staged tile
        const float (*cur)[KC + 4] = lb[ci & 1];
#pragma unroll
        for (int kk = 0; kk < KC; kk += 4) {
            v2f b = *(const v2f*)&cur[lane & 15][kk + ((lane >> 4) << 1)];
#pragma unroll
            for (int s = 0; s < 4; ++s) {
                v2f a = load_a(x_emb, HDIM, m0 + s * 16, k0 + kk, lane);
                acc[s] = wmma_f32(a, b, acc[s]);
            }
        }
    }
    const int n = n0 + (lane & 15);
    const float bias = b_ih[n];
#pragma unroll
    for (int s = 0; s < 4; ++s) {
        const int mbase = m0 + s * 16 + 8 * (lane >> 4);
#pragma unroll
        for (int r = 0; r < 8; ++r)
            gx[(size_t)(mbase + r) * GDIM + n] = acc[s][r] + bias;
    }
}

// ---------------------------------------------------------------------------
// lstm step: gates = gx_t + (h*keep) @ W_hh^T + b_hh; LSTM elementwise update.
// Block = 8 waves (256 thr) sharing one n-tile; wave tile = 32 rows x 16 cols
// x 4 gates = 8 accumulators. All four gate weight tiles async-staged in LDS,
// double buffered. Grid = C/16 = 32 blocks covering all 256 batch rows.
// ---------------------------------------------------------------------------
__global__ void __launch_bounds__(256)
lstm_step_kernel(const float* __restrict__ gx_t,   // [B, 4C] (has b_ih)
                 const float* __restrict__ h_in,   // [B, C]
                 float* __restrict__ c_io,         // [B, C] in/out (tile-local)
                 float* __restrict__ h_out,        // [B, C]
                 const float* __restrict__ W_hh,   // [4C, C]
                 const float* __restrict__ b_hh,   // [4C]
                 const float* __restrict__ mask_t) // [B]
{
    __shared__ float lb[2][4][16][KC + 4];
    const int wave = threadIdx.x >> 5;          // 0..7
    const int lane = threadIdx.x & 31;
    const int n0 = blockIdx.x * 16;
    const int m0 = wave * 32;

    // staging map: 4 x 16 x 64 = 4096 floats / 256 thr = 16 floats (4 x b128)
    const int row = threadIdx.x >> 2;            // 0..63 = gate*16 + n
    const int sq = row >> 4, sn = row & 15;
    const int kc = (threadIdx.x & 3) * 16;
    const float* wsrc = W_hh + (size_t)(sq * CDIM + n0 + sn) * CDIM + kc;
    float* ld[2] = { &lb[0][sq][sn][kc], &lb[1][sq][sn][kc] };

    v8f acc[4][2];
#pragma unroll
    for (int q = 0; q < 4; ++q)
#pragma unroll
        for (int s = 0; s < 2; ++s) acc[q][s] = (v8f){0.f,0.f,0.f,0.f,0.f,0.f,0.f,0.f};

    const float keep0 = 1.0f - mask_t[m0 + (lane & 15)];
    const float keep1 = 1.0f - mask_t[m0 + 16 + (lane & 15)];

    // prologue: stage chunk 0 into buffer 0
#pragma unroll
    for (int j = 0; j < 4; ++j) async_b128(wsrc + j * 4, ld[0] + j * 4);

    const int NCH = CDIM / KC;   // 8
    for (int ci = 0; ci < NCH; ++ci) {
        const int k0 = ci * KC;
        __syncthreads();
        if (ci + 1 < NCH) {
            const float* s = wsrc + (size_t)(ci + 1) * KC;
            float* d = ld[(ci + 1) & 1];
#pragma unroll
            for (int j = 0; j < 4; ++j) async_b128(s + j * 4, d + j * 4);
            ASYNC_WAIT(0x4);
        } else {
            ASYNC_WAIT(0x0);
        }
        __syncthreads();
        const float (*cur)[16][KC + 4] = lb[ci & 1];
#pragma unroll
        for (int kk = 0; kk < KC; kk += 4) {
            v2f a0 = load_a(h_in, CDIM, m0, k0 + kk, lane);
            a0.x *= keep0; a0.y *= keep0;
            v2f a1 = load_a(h_in, CDIM, m0 + 16, k0 + kk, lane);
            a1.x *= keep1; a1.y *= keep1;
#pragma unroll
            for (int q = 0; q < 4; ++q) {
                v2f b = *(const v2f*)&cur[q][lane & 15][kk + ((lane >> 4) << 1)];
                acc[q][0] = wmma_f32(a0, b, acc[q][0]);
                acc[q][1] = wmma_f32(a1, b, acc[q][1]);
            }
        }
    }

    const int n = n0 + (lane & 15);
    const float bi = b_hh[0 * CDIM + n];
    const float bf = b_hh[1 * CDIM + n];
    const float bg = b_hh[2 * CDIM + n];
    const float bo = b_hh[3 * CDIM + n];
#pragma unroll
    for (int s = 0; s < 2; ++s) {
        const int mbase = m0 + s * 16 + 8 * (lane >> 4);
#pragma unroll
        for (int r = 0; r < 8; ++r) {
            const int m = mbase + r;
            const float keep = 1.0f - mask_t[m];
            const size_t go = (size_t)m * GDIM + n;
            const float iv = sigf(acc[0][s][r] + gx_t[go + 0 * CDIM] + bi);
            const float fv = sigf(acc[1][s][r] + gx_t[go + 1 * CDIM] + bf);
            const float gv = tanhf(acc[2][s][r] + gx_t[go + 2 * CDIM] + bg);
            const float ov = sigf(acc[3][s][r] + gx_t[go + 3 * CDIM] + bo);
            const size_t off = (size_t)m * CDIM + n;
            const float cnew = fv * (c_io[off] * keep) + iv * gv;
            c_io[off]  = cnew;
            h_out[off] = ov * tanhf(cnew);
        }
    }
}

// ---------------------------------------------------------------------------
// mean head over a whole chunk: means = tanh(hiddens @ W_mean^T + b_mean).
// N=8 is too narrow for WMMA; rows = Tc*B gives good VALU occupancy.
// ---------------------------------------------------------------------------
__global__ void mean_kernel(const float* __restrict__ hid,     // [rows, C]
                            const float* __restrict__ W_mean,  // [ACT, C]
                            const float* __restrict__ b_mean,  // [ACT]
                            float* __restrict__ means,         // [rows, ACT]
                            int rows) {
    int idx = blockIdx.x * blockDim.x + threadIdx.x;
    if (idx >= rows * ACTD) return;
    const int rrow = idx >> 3, a = idx & 7;
    const float* hr = hid + (size_t)rrow * CDIM;
    const float* wr = W_mean + (size_t)a * CDIM;
    float s = b_mean[a];
#pragma unroll 8
    for (int k = 0; k < CDIM; ++k) s = fmaf(hr[k], wr[k], s);
    means[idx] = tanhf(s);
}

__global__ void copy_bc_kernel(const float* __restrict__ src, float* __restrict__ dst) {
    int idx = blockIdx.x * blockDim.x + threadIdx.x;
    if (idx < BATCH * CDIM) dst[idx] = src[idx];
}

__global__ void final_kernel(const float* __restrict__ hfin, const float* __restrict__ cfin,
                             float* __restrict__ out_hT, float* __restrict__ out_cT) {
    int idx = blockIdx.x * blockDim.x + threadIdx.x;
    if (idx < BATCH * CDIM) { out_hT[idx] = hfin[idx]; out_cT[idx] = cfin[idx]; }
}

// ---------------------------------------------------------------------------
extern "C" void kernel_launch(void* const* d_in, const int* in_sizes, int n_in,
                              void* d_out, int out_size, void* d_ws, size_t ws_size,
                              hipStream_t stream) {
    const float* xs      = (const float*)d_in[0];
    const float* h0      = (const float*)d_in[1];
    const float* c0      = (const float*)d_in[2];
    const float* h_masks = (const float*)d_in[3];
    const float* W_in    = (const float*)d_in[4];
    const float* b_in    = (const float*)d_in[5];
    const float* W_ih    = (const float*)d_in[6];
    const float* b_ih    = (const float*)d_in[7];
    const float* W_hh    = (const float*)d_in[8];
    const float* b_hh    = (const float*)d_in[9];
    const float* W_mean  = (const float*)d_in[10];
    const float* b_mean  = (const float*)d_in[11];
    const float* log_std = (const float*)d_in[12];

    // Pick chunk size Tc (power-of-two divisor of 512) to fit workspace:
    //   per-step floats: B*(H + 4C + C); fixed: 2*B*C (hcarry + cbuf)
    const size_t perT  = (size_t)BATCH * (HDIM + GDIM + CDIM);
    const size_t fixed = 2 * (size_t)BATCH * CDIM;
    int Tc = 1;
    for (int cand = 512; cand >= 1; cand >>= 1) {
        if ((fixed + perT * (size_t)cand) * sizeof(float) <= ws_size) { Tc = cand; break; }
    }

    float* ws      = (float*)d_ws;
    float* x_emb   = ws;                                   // [Tc*B, H]
    float* gx      = x_emb + (size_t)Tc * BATCH * HDIM;    // [Tc*B, 4C]
    float* hiddens = gx + (size_t)Tc * BATCH * GDIM;       // [Tc*B, C]
    float* hcarry  = hiddens + (size_t)Tc * BATCH * CDIM;  // [B, C]
    float* cbuf    = hcarry + (size_t)BATCH * CDIM;        // [B, C]

    float* out        = (float*)d_out;
    float* out_means  = out;
    float* out_logstd = out + (size_t)T_STEPS * BATCH * ACTD;
    float* out_hT     = out + 2 * (size_t)T_STEPS * BATCH * ACTD;
    float* out_cT     = out_hT + (size_t)BATCH * CDIM;

    init_kernel<<<(T_STEPS * BATCH * ACTD + 255) / 256, 256, 0, stream>>>(
        h0, c0, log_std, hcarry, cbuf, out_logstd);

    for (int t0 = 0; t0 < T_STEPS; t0 += Tc) {
        const int M = Tc * BATCH;

        embed_kernel<<<dim3(HDIM / 64, M / 16), 128, 0, stream>>>(
            xs + (size_t)t0 * BATCH * OBSD, W_in, b_in, x_emb);

        gx_kernel<<<dim3(GDIM / 16, M / 256), 128, 0, stream>>>(
            x_emb, W_ih, b_ih, gx);

        for (int tl = 0; tl < Tc; ++tl) {
            const int t = t0 + tl;
            const float* h_in = (tl == 0) ? hcarry
                                          : hiddens + (size_t)(tl - 1) * BATCH * CDIM;
            float* h_out = hiddens + (size_t)tl * BATCH * CDIM;
            lstm_step_kernel<<<CDIM / 16, 256, 0, stream>>>(
                gx + (size_t)tl * BATCH * GDIM, h_in, cbuf, h_out,
                W_hh, b_hh, h_masks + (size_t)t * BATCH);
        }

        mean_kernel<<<(M * ACTD + 255) / 256, 256, 0, stream>>>(
            hiddens, W_mean, b_mean, out_means + (size_t)t0 * BATCH * ACTD, M);

        copy_bc_kernel<<<(BATCH * CDIM + 255) / 256, 256, 0, stream>>>(
            hiddens + (size_t)(Tc - 1) * BATCH * CDIM, hcarry);
    }

    final_kernel<<<(BATCH * CDIM + 255) / 256, 256, 0, stream>>>(
        hcarry, cbuf, out_hT, out_cT);
}